// DynamicEmbedder_90950227460855
// MI455X (gfx1250) — compile-verified
//
#include <hip/hip_runtime.h>

#define BATCH 4
#define NPTS  200000
#define GRIDD 512
#define HW    (GRIDD * GRIDD)      // 262144 = 2^18
#define FEAT  64
#define NTILES (NPTS / 16)         // 12500

typedef __attribute__((ext_vector_type(2))) float v2f;
typedef __attribute__((ext_vector_type(8))) float v8f;

// ---------------- Pass 1: voxelize, pillar count + xyz sums ----------------
__global__ void voxelize_kernel(const float* __restrict__ pts,
                                float* __restrict__ cnt,
                                float* __restrict__ sumxyz,
                                int* __restrict__ segArr) {
  int idx = blockIdx.x * blockDim.x + threadIdx.x;
  if (idx >= BATCH * NPTS) return;
  int b = idx / NPTS;
  float x = pts[3 * idx + 0];
  float y = pts[3 * idx + 1];
  float z = pts[3 * idx + 2];
  int ix = (int)floorf((x + 51.2f) / 0.2f);
  int iy = (int)floorf((y + 51.2f) / 0.2f);
  ix = min(max(ix, 0), GRIDD - 1);
  iy = min(max(iy, 0), GRIDD - 1);
  int seg = iy * GRIDD + ix;
  segArr[idx] = seg;
  int sb = b * HW + seg;
  atomicAdd(&cnt[sb], 1.0f);
  atomicAdd(&sumxyz[3 * sb + 0], x);
  atomicAdd(&sumxyz[3 * sb + 1], y);
  atomicAdd(&sumxyz[3 * sb + 2], z);
}

// ---------------- Pass 2: feature build + [16x9]@[9x64] via WMMA f32 -------
__global__ void pfn_wmma_kernel(const float* __restrict__ pts,
                                const float* __restrict__ Wlin,
                                const float* __restrict__ blin,
                                const float* __restrict__ cnt,
                                const float* __restrict__ sumxyz,
                                float* __restrict__ hout,
                                float* __restrict__ csum,
                                float* __restrict__ csumsq) {
  const int b    = blockIdx.y;
  const int lane = threadIdx.x & 31;
  const int half = lane >> 4;   // 0: K 0-1 / rows 0-7 ; 1: K 2-3 / rows 8-15
  const int nl   = lane & 15;
  const int wave   = (blockIdx.x * blockDim.x + threadIdx.x) >> 5;
  const int nwaves = (gridDim.x * blockDim.x) >> 5;

  // B-matrix fragments: 3 K-tiles (K padded 9->12) x 4 N-tiles of 16 channels
  // layout: lanes 0-15 hold K=4t..4t+1 (vgpr0/1), lanes 16-31 hold K=4t+2..4t+3
  v2f bf[3][4];
  float bias[4];
#pragma unroll
  for (int t = 0; t < 3; ++t) {
#pragma unroll
    for (int q = 0; q < 4; ++q) {
      int k0 = 4 * t + 2 * half;
      int n  = q * 16 + nl;
      bf[t][q][0] = (k0 < 9)     ? Wlin[k0 * FEAT + n]       : 0.0f;
      bf[t][q][1] = (k0 + 1 < 9) ? Wlin[(k0 + 1) * FEAT + n] : 0.0f;
    }
  }
#pragma unroll
  for (int q = 0; q < 4; ++q) bias[q] = blin[q * 16 + nl];

  float cs[4]  = {0.f, 0.f, 0.f, 0.f};
  float cq2[4] = {0.f, 0.f, 0.f, 0.f};

  for (int t = wave; t < NTILES; t += nwaves) {
    const int pt0 = t * 16;
    const int p   = b * NPTS + pt0 + nl;  // both halves load the same point
    float x = pts[3 * p + 0];
    float y = pts[3 * p + 1];
    float z = pts[3 * p + 2];
    int ix = (int)floorf((x + 51.2f) / 0.2f);
    int iy = (int)floorf((y + 51.2f) / 0.2f);
    ix = min(max(ix, 0), GRIDD - 1);
    iy = min(max(iy, 0), GRIDD - 1);
    int seg = iy * GRIDD + ix;
    int sb  = b * HW + seg;
    float inv = 1.0f / fmaxf(cnt[sb], 1.0f);
    float mx = sumxyz[3 * sb + 0] * inv;
    float my = sumxyz[3 * sb + 1] * inv;
    float mz = sumxyz[3 * sb + 2] * inv;
    float ccx = ((float)ix + 0.5f) * 0.2f - 51.2f;
    float ccy = ((float)iy + 0.5f) * 0.2f - 51.2f;
    // feats: [x,y,z, x-mx,y-my,z-mz, x-cx,y-cy,z-cz]; cz = -3 + 0.5*6 = 0
    float f0 = x, f1 = y, f2 = z;
    float f3 = x - mx, f4 = y - my, f5 = z - mz;
    float f6 = x - ccx, f7 = y - ccy, f8 = z;

    // A fragments 16x4 per K-tile (K 9..11 zero-padded)
    v2f a0, a1, a2;
    a0[0] = half ? f2 : f0;   a0[1] = half ? f3 : f1;
    a1[0] = half ? f6 : f4;   a1[1] = half ? f7 : f5;
    a2[0] = half ? 0.0f : f8; a2[1] = 0.0f;

#pragma unroll
    for (int q = 0; q < 4; ++q) {
      v8f acc = {};
      acc = __builtin_amdgcn_wmma_f32_16x16x4_f32(false, a0, false, bf[0][q],
                                                  (short)0, acc, false, false);
      acc = __builtin_amdgcn_wmma_f32_16x16x4_f32(false, a1, false, bf[1][q],
                                                  (short)0, acc, false, false);
      acc = __builtin_amdgcn_wmma_f32_16x16x4_f32(false, a2, false, bf[2][q],
                                                  (short)0, acc, false, false);
#pragma unroll
      for (int r = 0; r < 8; ++r) {
        float v = acc[r] + bias[q];                  // D vgpr r: rows r, r+8
        int row = pt0 + r + 8 * half;
        hout[(size_t)(b * NPTS + row) * FEAT + q * 16 + nl] = v;  // raw h
        cs[q]  += v;
        cq2[q] += v * v;
      }
    }
  }

  // per-channel sums: fold the two half-wave copies, then a few atomics
#pragma unroll
  for (int q = 0; q < 4; ++q) {
    float s  = cs[q]  + __shfl_xor(cs[q], 16, 32);
    float s2 = cq2[q] + __shfl_xor(cq2[q], 16, 32);
    if (half == 0) {
      atomicAdd(&csum[b * FEAT + q * 16 + nl], s);
      atomicAdd(&csumsq[b * FEAT + q * 16 + nl], s2);
    }
  }
}

// ---------------- Pass 3: batch-norm statistics -> scale/shift --------------
__global__ void stats_kernel(const float* __restrict__ csum,
                             const float* __restrict__ csumsq,
                             const float* __restrict__ gamma,
                             const float* __restrict__ beta,
                             float* __restrict__ scale,
                             float* __restrict__ shift) {
  int idx = threadIdx.x;  // BATCH*FEAT == 256
  int c = idx & (FEAT - 1);
  const float invN = 1.0f / (float)NPTS;
  float mu  = csum[idx] * invN;
  float var = csumsq[idx] * invN - mu * mu;   // population variance (ddof=0)
  float sc  = gamma[c] * rsqrtf(var + 1e-5f);
  scale[idx] = sc;
  shift[idx] = beta[c] - mu * sc;
}

// ---------------- Pass 4: BN + ReLU in place, scatter into pseudo-image -----
__global__ void bn_scatter_kernel(float* __restrict__ hio,
                                  const float* __restrict__ scale,
                                  const float* __restrict__ shift,
                                  const int* __restrict__ segArr,
                                  float* __restrict__ pseudo) {
  int idx = blockIdx.x * blockDim.x + threadIdx.x;  // < BATCH*NPTS*FEAT
  int c = idx & (FEAT - 1);
  int p = idx >> 6;          // global point id
  int b = p / NPTS;
  float v = hio[idx];
  v = fmaxf(fmaf(v, scale[b * FEAT + c], shift[b * FEAT + c]), 0.0f);
  hio[idx] = v;              // final point_feats
  int seg = segArr[p];
  atomicAdd(&pseudo[((size_t)(b * FEAT + c)) * HW + seg], v);
}

// ---------------- Pass 5: divide pseudo-image by pillar counts --------------
__global__ void finalize_kernel(float* __restrict__ pseudo,
                                const float* __restrict__ cnt) {
  int idx = blockIdx.x * blockDim.x + threadIdx.x;  // < BATCH*FEAT*HW
  int p = idx & (HW - 1);
  int b = idx >> 24;  // FEAT*HW == 2^24
  float c = cnt[b * HW + p];
  pseudo[idx] *= 1.0f / fmaxf(c, 1.0f);
}

extern "C" void kernel_launch(void* const* d_in, const int* in_sizes, int n_in,
                              void* d_out, int out_size, void* d_ws, size_t ws_size,
                              hipStream_t stream) {
  (void)in_sizes; (void)n_in; (void)out_size; (void)ws_size;
  const float* pts   = (const float*)d_in[0];
  const float* Wlin  = (const float*)d_in[1];
  const float* blin  = (const float*)d_in[2];
  const float* gamma = (const float*)d_in[3];
  const float* beta  = (const float*)d_in[4];

  float* out    = (float*)d_out;
  float* pseudo = out;                                  // [B,64,512,512]
  float* hout   = out + (size_t)BATCH * FEAT * HW;      // [B,N,64]

  // scratch layout in d_ws (~19.1 MiB)
  char* ws = (char*)d_ws;
  float* cnt    = (float*)(ws + 0);          // B*HW floats        (4 MiB)
  float* sumxyz = (float*)(ws + 4194304);    // B*HW*3 floats      (12 MiB)
  float* csum   = (float*)(ws + 16777216);   // B*64
  float* csumsq = (float*)(ws + 16778240);   // B*64
  float* scale  = (float*)(ws + 16779264);   // B*64
  float* shift  = (float*)(ws + 16780288);   // B*64
  int*   segArr = (int*)  (ws + 16781312);   // B*N ints           (3.2 MB)

  hipMemsetAsync(pseudo, 0, (size_t)BATCH * FEAT * HW * sizeof(float), stream);
  hipMemsetAsync(ws, 0, 16779264 + 2048, stream);  // cnt+sumxyz+csum+csumsq+scale+shift

  voxelize_kernel<<<(BATCH * NPTS + 255) / 256, 256, 0, stream>>>(
      pts, cnt, sumxyz, segArr);
  pfn_wmma_kernel<<<dim3(64, BATCH), 256, 0, stream>>>(
      pts, Wlin, blin, cnt, sumxyz, hout, csum, csumsq);
  stats_kernel<<<1, 256, 0, stream>>>(csum, csumsq, gamma, beta, scale, shift);
  bn_scatter_kernel<<<(BATCH * NPTS * FEAT) / 256, 256, 0, stream>>>(
      hout, scale, shift, segArr, pseudo);
  finalize_kernel<<<(BATCH * FEAT * HW) / 256, 256, 0, stream>>>(pseudo, cnt);
}